// Social_Aggregator_25821343383579
// MI455X (gfx1250) — compile-verified
//
#include <hip/hip_runtime.h>
#include <math.h>

typedef float v2f __attribute__((ext_vector_type(2)));
typedef float v8f __attribute__((ext_vector_type(8)));

#define DD 64
#define TWO_D 128
#define WAVES 4
#define TILE 16     // edge rows per wave tile (matches WMMA M=16)
#define XP 132      // padded LDS pitch for 128-wide rows (bank-conflict-free frags)
#define X2P 68      // padded LDS pitch for 64-wide rows
#define W1P 132
#define W2P 68

// ---------------- Kernel 1: per-edge MLP -> logits, fp32 WMMA ----------------
__global__ __launch_bounds__(WAVES * 32)
void edge_mlp_kernel(const int* __restrict__ nodes, const int* __restrict__ neigh_idx,
                     const int* __restrict__ seg_ids, const float* __restrict__ u2e,
                     const float* __restrict__ att1_w, const float* __restrict__ att1_b,
                     const float* __restrict__ att2_w, const float* __restrict__ att2_b,
                     const float* __restrict__ att3_w, const float* __restrict__ att3_b,
                     float* __restrict__ logits, int E)
{
    __shared__ float WT1[DD][W1P];          // att1_w transposed: WT1[n][k], k<128
    __shared__ float WT2[DD][W2P];          // att2_w transposed: WT2[n][k], k<64
    __shared__ float B1s[DD], B2s[DD], W3s[DD];
    __shared__ float B3s;
    __shared__ float X [WAVES][TILE][XP];   // layer-1 input rows [e_u | u_rep], also reused for layer-2 output
    __shared__ float X2[WAVES][TILE][X2P];  // layer-1 output rows

    const int tid  = threadIdx.x;
    const int lane = tid & 31;
    const int wv   = tid >> 5;
    const int t    = lane & 15;   // N (or M for A frags) within tile
    const int hi   = lane >> 4;   // half-wave select (K split)

    // ---- stage weights (transposed) + biases into LDS ----
    for (int i = tid; i < TWO_D * DD; i += WAVES * 32) {
        int k = i >> 6, n = i & 63;
        WT1[n][k] = att1_w[i];
    }
    for (int i = tid; i < DD * DD; i += WAVES * 32) {
        int k = i >> 6, n = i & 63;
        WT2[n][k] = att2_w[i];
    }
    if (tid < DD) { B1s[tid] = att1_b[tid]; B2s[tid] = att2_b[tid]; W3s[tid] = att3_w[tid]; }
    if (tid == 0) B3s = att3_b[0];
    __syncthreads();

    const int tile = blockIdx.x * WAVES + wv;
    const int e0   = tile * TILE;

    // ---- gather X rows: lanes 0-15 -> e_u (cols 0..63), lanes 16-31 -> u_rep (cols 64..127)
    for (int r = 0; r < TILE; ++r) {
        int e = e0 + r; if (e >= E) e = E - 1;
        int src_row = (lane < 16) ? neigh_idx[e] : nodes[seg_ids[e]];
        const float* src = u2e + (size_t)src_row * DD + t * 4;
        float4 v = *(const float4*)src;
        float* dst = &X[wv][r][(lane < 16 ? 0 : DD) + t * 4];
        dst[0] = v.x; dst[1] = v.y; dst[2] = v.z; dst[3] = v.w;
    }
    __syncthreads();

    // ---- layer 1: [16x128] x [128x64] via V_WMMA_F32_16X16X4_F32, K stepped by 4 ----
    v8f acc[4];
    #pragma unroll
    for (int nt = 0; nt < 4; ++nt) acc[nt] = (v8f){0,0,0,0,0,0,0,0};

    for (int kk = 0; kk < TWO_D; kk += 4) {
        v2f a;                                 // A 16x4: VGPR0=K0/K2, VGPR1=K1/K3 per half-wave
        a.x = X[wv][t][kk + 2 * hi];
        a.y = X[wv][t][kk + 2 * hi + 1];
        #pragma unroll
        for (int nt = 0; nt < 4; ++nt) {
            v2f b;                             // B 4x16: row K striped over lanes (N = t)
            b.x = WT1[nt * 16 + t][kk + 2 * hi];
            b.y = WT1[nt * 16 + t][kk + 2 * hi + 1];
            acc[nt] = __builtin_amdgcn_wmma_f32_16x16x4_f32(
                false, a, false, b, (short)0, acc[nt], false, false);
        }
    }
    // bias + relu + writeback (D layout: VGPR r -> M = r + 8*hi, N = nt*16 + t)
    #pragma unroll
    for (int nt = 0; nt < 4; ++nt) {
        int n = nt * 16 + t;
        float bias = B1s[n];
        #pragma unroll
        for (int r = 0; r < 8; ++r) {
            float val = fmaxf(acc[nt][r] + bias, 0.0f);
            X2[wv][r + 8 * hi][n] = val;
        }
    }
    __syncthreads();

    // ---- layer 2: [16x64] x [64x64] ----
    #pragma unroll
    for (int nt = 0; nt < 4; ++nt) acc[nt] = (v8f){0,0,0,0,0,0,0,0};

    for (int kk = 0; kk < DD; kk += 4) {
        v2f a;
        a.x = X2[wv][t][kk + 2 * hi];
        a.y = X2[wv][t][kk + 2 * hi + 1];
        #pragma unroll
        for (int nt = 0; nt < 4; ++nt) {
            v2f b;
            b.x = WT2[nt * 16 + t][kk + 2 * hi];
            b.y = WT2[nt * 16 + t][kk + 2 * hi + 1];
            acc[nt] = __builtin_amdgcn_wmma_f32_16x16x4_f32(
                false, a, false, b, (short)0, acc[nt], false, false);
        }
    }
    #pragma unroll
    for (int nt = 0; nt < 4; ++nt) {
        int n = nt * 16 + t;
        float bias = B2s[n];
        #pragma unroll
        for (int r = 0; r < 8; ++r) {
            float val = fmaxf(acc[nt][r] + bias, 0.0f);
            X[wv][r + 8 * hi][n] = val;   // reuse X as layer-2 output tile
        }
    }
    __syncthreads();

    // ---- logits: row dot with att3_w (lanes 0-15 own one row each) ----
    if (lane < 16) {
        float lg = B3s;
        #pragma unroll
        for (int n = 0; n < DD; ++n) lg += X[wv][lane][n] * W3s[n];
        int e = e0 + lane;
        if (e < E) logits[e] = lg;
    }
}

// ---------------- Kernel 2: segment softmax + weighted aggregation ----------------
__device__ __forceinline__ int lbound(const int* __restrict__ a, int n, int key) {
    int lo = 0, hi = n;
    while (lo < hi) {
        int mid = (lo + hi) >> 1;
        if (a[mid] < key) lo = mid + 1; else hi = mid;
    }
    return lo;
}

__global__ __launch_bounds__(256)
void seg_softmax_agg_kernel(const int* __restrict__ neigh_idx, const int* __restrict__ seg_ids,
                            const float* __restrict__ u2e, const float* __restrict__ logits,
                            float* __restrict__ out, int E, int N)
{
    const int lane = threadIdx.x & 31;
    const int node = blockIdx.x * (blockDim.x >> 5) + (threadIdx.x >> 5);
    if (node >= N) return;                       // wave-uniform

    const int start = lbound(seg_ids, E, node);
    const int end   = lbound(seg_ids, E, node + 1);

    if (start >= end) {                          // empty segment -> zeros
        out[(size_t)node * DD + lane]      = 0.0f;
        out[(size_t)node * DD + lane + 32] = 0.0f;
        return;
    }

    // pass 1: segment max
    float m = -INFINITY;
    for (int e = start + lane; e < end; e += 32) m = fmaxf(m, logits[e]);
    for (int off = 16; off; off >>= 1) m = fmaxf(m, __shfl_xor(m, off, 32));

    // pass 2: segment sum of exp
    float s = 0.0f;
    for (int e = start + lane; e < end; e += 32) s += expf(logits[e] - m);
    for (int off = 16; off; off >>= 1) s += __shfl_xor(s, off, 32);
    const float inv = 1.0f / fmaxf(s, 1e-9f);

    // pass 3: weighted sum of neighbor embeddings; lane owns cols {lane, lane+32}
    float a0 = 0.0f, a1 = 0.0f;
    for (int base = start; base < end; base += 32) {
        int e = base + lane;
        float w = 0.0f; int idx = 0;
        if (e < end) { w = expf(logits[e] - m) * inv; idx = neigh_idx[e]; }
        int cnt = min(32, end - base);
        for (int j = 0; j < cnt; ++j) {
            float wj = __shfl(w, j, 32);
            int   ij = __shfl(idx, j, 32);
            const float* row = u2e + (size_t)ij * DD;
            a0 += wj * row[lane];
            a1 += wj * row[lane + 32];
        }
    }
    out[(size_t)node * DD + lane]      = a0;
    out[(size_t)node * DD + lane + 32] = a1;
}

// ---------------- launch ----------------
extern "C" void kernel_launch(void* const* d_in, const int* in_sizes, int n_in,
                              void* d_out, int out_size, void* d_ws, size_t ws_size,
                              hipStream_t stream) {
    const int*   nodes     = (const int*)d_in[0];
    const int*   neigh_idx = (const int*)d_in[1];
    const int*   seg_ids   = (const int*)d_in[2];
    const float* u2e       = (const float*)d_in[3];
    const float* att1_w    = (const float*)d_in[4];
    const float* att1_b    = (const float*)d_in[5];
    const float* att2_w    = (const float*)d_in[6];
    const float* att2_b    = (const float*)d_in[7];
    const float* att3_w    = (const float*)d_in[8];
    const float* att3_b    = (const float*)d_in[9];

    const int E = in_sizes[1];
    const int N = out_size / DD;

    float* logits = (float*)d_ws;           // E floats (2.56 MB) in scratch
    float* out    = (float*)d_out;

    const int tiles   = (E + TILE - 1) / TILE;
    const int blocks1 = (tiles + WAVES - 1) / WAVES;
    edge_mlp_kernel<<<blocks1, WAVES * 32, 0, stream>>>(
        nodes, neigh_idx, seg_ids, u2e, att1_w, att1_b, att2_w, att2_b,
        att3_w, att3_b, logits, E);

    const int wpb     = 8;                  // waves (nodes) per block
    const int blocks2 = (N + wpb - 1) / wpb;
    seg_softmax_agg_kernel<<<blocks2, wpb * 32, 0, stream>>>(
        neigh_idx, seg_ids, u2e, logits, out, E, N);
}